// Hyp_MLR_70549132804741
// MI455X (gfx1250) — compile-verified
//
#include <hip/hip_runtime.h>
#include <hip/hip_bf16.h>

typedef __attribute__((ext_vector_type(2))) float v2f;
typedef __attribute__((ext_vector_type(8))) float v8f;

#define N_PTS 4096
#define C_CLS 256
#define D_DIM 128
#define CURVF 1.0f

// ---------------- per-class precompute: p2[c], pa[c] = -(p·a)/max(||a||,eps), na[c] = ||a|| ----
__global__ __launch_bounds__(256) void hyp_pre_c(const float* __restrict__ a,
                                                 const float* __restrict__ p,
                                                 float* __restrict__ p2v,
                                                 float* __restrict__ pav,
                                                 float* __restrict__ nav) {
    int c = blockIdx.x * blockDim.x + threadIdx.x;
    if (c >= C_CLS) return;
    const float4* pr = (const float4*)(p + c * D_DIM);
    const float4* ar = (const float4*)(a + c * D_DIM);
    float sp2 = 0.f, spa = 0.f, sa2 = 0.f;
#pragma unroll 8
    for (int i = 0; i < D_DIM / 4; ++i) {
        float4 pv = pr[i], av = ar[i];
        sp2 += pv.x * pv.x + pv.y * pv.y + pv.z * pv.z + pv.w * pv.w;
        spa += pv.x * av.x + pv.y * av.y + pv.z * av.z + pv.w * av.w;
        sa2 += av.x * av.x + av.y * av.y + av.z * av.z + av.w * av.w;
    }
    float na = sqrtf(sa2);
    float nac = fmaxf(na, 1e-12f);
    p2v[c] = sp2;
    pav[c] = -spa / nac;
    nav[c] = na;
}

// ---------------- per-point precompute: x2[n] ----------------
__global__ __launch_bounds__(256) void hyp_pre_n(const float* __restrict__ x,
                                                 float* __restrict__ x2v) {
    int n = blockIdx.x * blockDim.x + threadIdx.x;
    const float4* xr = (const float4*)(x + n * D_DIM);
    float s = 0.f;
#pragma unroll 8
    for (int i = 0; i < D_DIM / 4; ++i) {
        float4 xv = xr[i];
        s += xv.x * xv.x + xv.y * xv.y + xv.z * xv.z + xv.w * xv.w;
    }
    x2v[n] = s;
}

// ---------------- fused dual-GEMM (f32 WMMA) + hyperbolic epilogue ----------------
// One wave32 computes a 16(n) x 16(c) output tile.
//   acc_py = X_tile (16xK) x P_tile^T (Kx16)   -> p·x  (xy = -acc_py)
//   acc_xa = X_tile (16xK) x A_tile^T (Kx16)   -> a·x  (xa = acc_xa / ||a||)
// F32 WMMA 16x16x4 fragment addressing (row-major source, row length D):
//   frag[j] = src[(base + lane%16)*D + k + 2*(lane>>4) + j],  j in {0,1}
__global__ __launch_bounds__(256) void hyp_mlr_main(const float* __restrict__ X,
                                                    const float* __restrict__ A,
                                                    const float* __restrict__ P,
                                                    const float* __restrict__ p2v,
                                                    const float* __restrict__ pav,
                                                    const float* __restrict__ nav,
                                                    const float* __restrict__ x2v,
                                                    float* __restrict__ out) {
    const int lane = threadIdx.x & 31;
    const int wave = threadIdx.x >> 5;
    const int tile = blockIdx.x * 8 + wave;        // 4096 tiles total
    const int tN = tile & ((N_PTS / 16) - 1);      // 256 tiles along N
    const int tC = tile >> 8;                      // 16 tiles along C
    const int n0 = tN * 16;
    const int c0 = tC * 16;
    const int m = lane & 15;
    const int half = lane >> 4;

    const float* xrow = X + (size_t)(n0 + m) * D_DIM + 2 * half;
    const float* prow = P + (size_t)(c0 + m) * D_DIM + 2 * half;
    const float* arow = A + (size_t)(c0 + m) * D_DIM + 2 * half;

    v8f acc_py = {};
    v8f acc_xa = {};
#pragma unroll
    for (int k = 0; k < D_DIM; k += 4) {
        v2f af = *(const v2f*)(xrow + k);
        v2f bp = *(const v2f*)(prow + k);
        v2f ba = *(const v2f*)(arow + k);
        // D = A*B + C ; 8-arg form: (neg_a, A, neg_b, B, c_mod, C, reuse_a, reuse_b)
        acc_py = __builtin_amdgcn_wmma_f32_16x16x4_f32(false, af, false, bp,
                                                       (short)0, acc_py, false, false);
        acc_xa = __builtin_amdgcn_wmma_f32_16x16x4_f32(false, af, false, ba,
                                                       (short)0, acc_xa, false, false);
    }

    const int cc = c0 + m;
    const float p2 = p2v[cc];
    const float pa = pav[cc];          // already -(p·a)/||a||
    const float na = nav[cc];
    const float inv_na = 1.0f / fmaxf(na, 1e-12f);
    const float cu = CURVF;

    // C/D layout: VGPR r holds row M = r + 8*half, column N = lane%16
#pragma unroll
    for (int r = 0; r < 8; ++r) {
        const int n = n0 + r + 8 * half;
        const float x2 = x2v[n];
        const float xy = -acc_py[r];               // (-p)·x
        const float xa = acc_xa[r] * inv_na;       // x·(a/||a||)
        const float t  = 1.0f + 2.0f * cu * xy;
        const float A1 = t + cu * x2;              // num_p scale
        const float B1 = 1.0f - cu * p2;           // num_x scale
        const float den = t + cu * cu * p2 * x2;
        const float inv_den = 1.0f / den;
        const float rr = (A1 * A1 * p2 + 2.0f * A1 * B1 * xy + B1 * B1 * x2)
                         * inv_den * inv_den;      // |res|^2
        const float lam = 2.0f / (1.0f - rr);
        const float dot = (A1 * pa + B1 * xa) * inv_den;
        out[(size_t)n * C_CLS + cc] = 2.0f * na * asinhf(dot * lam);
    }
}

extern "C" void kernel_launch(void* const* d_in, const int* in_sizes, int n_in,
                              void* d_out, int out_size, void* d_ws, size_t ws_size,
                              hipStream_t stream) {
    const float* X = (const float*)d_in[0];   // output_before (N, D)
    const float* A = (const float*)d_in[1];   // a_mlr (C, D)
    const float* P = (const float*)d_in[2];   // p_mlr (C, D)
    float* out = (float*)d_out;               // (N, C)

    float* ws  = (float*)d_ws;                // 4864 floats = 19.5 KB
    float* p2v = ws;
    float* pav = ws + C_CLS;
    float* nav = ws + 2 * C_CLS;
    float* x2v = ws + 3 * C_CLS;

    hyp_pre_c<<<1, 256, 0, stream>>>(A, P, p2v, pav, nav);
    hyp_pre_n<<<N_PTS / 256, 256, 0, stream>>>(X, x2v);

    const int tiles = (N_PTS / 16) * (C_CLS / 16);   // 4096
    hyp_mlr_main<<<tiles / 8, 256, 0, stream>>>(X, A, P, p2v, pav, nav, x2v, out);
}